// HungarianMatcher_53154515255431
// MI455X (gfx1250) — compile-verified
//
#include <hip/hip_runtime.h>
#include <hip/hip_bf16.h>

typedef __attribute__((ext_vector_type(2))) float v2f;
typedef __attribute__((ext_vector_type(8))) float v8f;

#define BS   16
#define Q    1024
#define NC   92
#define NCP  96     // padded classes (multiple of K=4)
#define T    128
#define INFV 3.4e38f

// ---------------------------------------------------------------------------
// Kernel 1: negated softmax, padded to 96 classes. One wave (32 lanes) per row.
// ---------------------------------------------------------------------------
__global__ void __launch_bounds__(128)
softmax_neg_kernel(const float* __restrict__ logits, float* __restrict__ negProb) {
    int row  = blockIdx.x * 4 + (threadIdx.x >> 5);   // 16384 rows total
    int lane = threadIdx.x & 31;
    const float* x = logits + (size_t)row * NC;

    float v0 = x[lane];                                   // lane      < 92 always
    float v1 = (lane + 32 < NC) ? x[lane + 32] : -INFV;
    float v2 = (lane + 64 < NC) ? x[lane + 64] : -INFV;

    float m = fmaxf(v0, fmaxf(v1, v2));
    for (int off = 16; off; off >>= 1) m = fmaxf(m, __shfl_xor(m, off, 32));

    float e0 = __expf(v0 - m);
    float e1 = (lane + 32 < NC) ? __expf(v1 - m) : 0.0f;
    float e2 = (lane + 64 < NC) ? __expf(v2 - m) : 0.0f;
    float s = e0 + e1 + e2;
    for (int off = 16; off; off >>= 1) s += __shfl_xor(s, off, 32);

    float ninv = -1.0f / s;  // store -prob so A@onehot directly yields cost_class
    float* o = negProb + (size_t)row * NCP;
    o[lane]      = e0 * ninv;
    o[lane + 32] = e1 * ninv;
    o[lane + 64] = e2 * ninv;  // lanes 28..31 write the zero padding (c=92..95)
}

// ---------------------------------------------------------------------------
// Kernel 2: cost matrix via WMMA f32 16x16x4 (one-hot matmul) + fused L1 cost.
// Grid: 256 blocks = (batch, 64-query block); 128 threads = 4 waves;
// each wave owns one 16-row M-tile and sweeps the 8 N-tiles (128 targets).
// ---------------------------------------------------------------------------
__global__ void __launch_bounds__(128)
cost_wmma_kernel(const float* __restrict__ negProb,
                 const float* __restrict__ pred_points,
                 const int*   __restrict__ tgt_labels,
                 const float* __restrict__ tgt_points,
                 float* __restrict__ Cb,          // (16,1024,128) -> d_out
                 float* __restrict__ CbT) {       // (16,128,1024) -> ws (optional)
    const int b    = blockIdx.x >> 4;
    const int qblk = blockIdx.x & 15;
    const int wave = threadIdx.x >> 5;
    const int lane = threadIdx.x & 31;
    const int tid  = threadIdx.x;

    __shared__ int   sLab[T];
    __shared__ float sTpt[2 * T];     // target (x,y)
    __shared__ float sQpt[2 * 64];    // this block's 64 query points

    if (tid < T) sLab[tid] = tgt_labels[b * T + tid];
    for (int k = tid; k < 2 * T; k += 128) sTpt[k] = tgt_points[(size_t)b * 2 * T + k];
    for (int k = tid; k < 128; k += 128)
        sQpt[k] = pred_points[((size_t)b * Q + qblk * 64) * 2 + k];
    __syncthreads();

    const int qbase = qblk * 64 + wave * 16;
    const int mrow  = lane & 15;
    const int khalf = (lane >> 4) * 2;  // lanes 0-15: K=0,1 ; lanes 16-31: K=2,3
    const float* Arow = negProb + ((size_t)(b * Q + qbase + mrow)) * NCP + khalf;

    const int n    = lane & 15;
    const int mofs = (lane >> 4) * 8;

    for (int nt = 0; nt < 8; ++nt) {
        const int tcol = nt * 16 + n;
        const int labn = sLab[tcol];

        v8f acc = {};
        #pragma unroll 4
        for (int k0 = 0; k0 < NCP; k0 += 4) {
            v2f a;  a.x = Arow[k0];  a.y = Arow[k0 + 1];   // A: 16x4 f32 fragment
            const int kk = k0 + khalf;                     // B: on-the-fly one-hot
            v2f bb; bb.x = (labn == kk)     ? 1.0f : 0.0f;
                    bb.y = (labn == kk + 1) ? 1.0f : 0.0f;
            acc = __builtin_amdgcn_wmma_f32_16x16x4_f32(
                false, a, false, bb, (short)0, acc, false, false);
        }

        const float tx = sTpt[2 * tcol], ty = sTpt[2 * tcol + 1];
        #pragma unroll
        for (int r = 0; r < 8; ++r) {
            const int m    = r + mofs;              // C layout: VGPR r -> M = r / 8+r
            const int qloc = wave * 16 + m;
            const float qx = sQpt[2 * qloc], qy = sQpt[2 * qloc + 1];
            const float c  = acc[r] + fabsf(qx - tx) + fabsf(qy - ty);
            const int  q   = qbase + m;
            Cb[((size_t)b * Q + q) * T + tcol] = c;
            if (CbT) CbT[((size_t)b * T + tcol) * Q + q] = c;   // transposed copy
        }
    }
}

// ---------------------------------------------------------------------------
// Kernel 3: Jonker-Volgenant on the transposed matrix (128 rows x 1024 cols).
// One workgroup per batch, 256 threads; all solver state in LDS.
// ---------------------------------------------------------------------------
__global__ void __launch_bounds__(256)
jv_kernel(const float* __restrict__ Cb, const float* __restrict__ CbT,
          float* __restrict__ d_out) {
    const int b = blockIdx.x, tid = threadIdx.x;
    const int NT = 256;

    __shared__ float v_[Q + 1];
    __shared__ float minv[Q + 1];
    __shared__ int   way[Q + 1];
    __shared__ int   p_[Q + 1];
    __shared__ unsigned char used[Q + 1];
    __shared__ float u_[T + 1];
    __shared__ float redv[256];
    __shared__ int   redj[256];
    __shared__ int   sh_j0, sh_i0, sh_done;
    __shared__ int   ans[T];

    const float* costT = CbT ? (CbT + (size_t)b * T * Q) : nullptr;  // row-major cost^T
    const float* costQ = Cb + (size_t)b * Q * T;                     // fallback, strided

    for (int j = tid; j <= Q; j += NT) { v_[j] = 0.0f; p_[j] = 0; }
    for (int i = tid; i <= T; i += NT) u_[i] = 0.0f;
    __syncthreads();

    for (int i = 1; i <= T; ++i) {
        if (tid == 0) { p_[0] = i; sh_j0 = 0; }
        for (int j = tid; j <= Q; j += NT) { minv[j] = INFV; way[j] = 0; used[j] = 0; }
        __syncthreads();

        while (true) {
            if (tid == 0) { used[sh_j0] = 1; sh_i0 = p_[sh_j0]; }
            __syncthreads();
            const int j0 = sh_j0, i0 = sh_i0;
            const float ui0 = u_[i0];

            // update minv/way over free columns; local argmin (first-index tiebreak)
            float bestv = INFV; int bestj = Q + 1;
            for (int j = tid + 1; j <= Q; j += NT) {
                if (!used[j]) {
                    const float cij = costT ? costT[(size_t)(i0 - 1) * Q + (j - 1)]
                                            : costQ[(size_t)(j - 1) * T + (i0 - 1)];
                    const float cur = cij - ui0 - v_[j];
                    float mv = minv[j];
                    if (cur < mv) { mv = cur; minv[j] = cur; way[j] = j0; }
                    if (mv < bestv || (mv == bestv && j < bestj)) { bestv = mv; bestj = j; }
                }
            }
            redv[tid] = bestv; redj[tid] = bestj;
            __syncthreads();
            for (int s = 128; s > 0; s >>= 1) {
                if (tid < s) {
                    const float ov = redv[tid + s]; const int oj = redj[tid + s];
                    if (ov < redv[tid] || (ov == redv[tid] && oj < redj[tid])) {
                        redv[tid] = ov; redj[tid] = oj;
                    }
                }
                __syncthreads();
            }
            const float delta = redv[0];
            const int   j1    = redj[0];

            // dual updates (distinct p_[j] across used columns -> race-free)
            for (int j = tid; j <= Q; j += NT) {
                if (used[j]) { u_[p_[j]] += delta; v_[j] -= delta; }
                else         { minv[j] -= delta; }
            }
            if (tid == 0) { sh_j0 = j1; sh_done = (p_[j1] == 0); }
            __syncthreads();
            if (sh_done) break;
        }

        if (tid == 0) {           // augmenting-path pointer chase (serial, short)
            int j0 = sh_j0;
            while (j0 != 0) { const int j1 = way[j0]; p_[j0] = p_[j1]; j0 = j1; }
        }
        __syncthreads();
    }

    // ans[target] = matched query
    for (int j = tid + 1; j <= Q; j += NT) if (p_[j] > 0) ans[p_[j] - 1] = j - 1;
    __syncthreads();

    // emulate argsort(ans): rank by assigned query (values are distinct)
    float* rowOut = d_out + (size_t)BS * Q * T + (size_t)b * T;
    float* colOut = rowOut + (size_t)BS * T;
    if (tid < T) {
        const int a = ans[tid];
        int r = 0;
        for (int k = 0; k < T; ++k) r += (ans[k] < a);
        rowOut[r] = (float)a;    // row_idx: sorted query indices
        colOut[r] = (float)tid;  // col_idx: matching target indices
    }
}

// ---------------------------------------------------------------------------
extern "C" void kernel_launch(void* const* d_in, const int* in_sizes, int n_in,
                              void* d_out, int out_size, void* d_ws, size_t ws_size,
                              hipStream_t stream) {
    const float* logits = (const float*)d_in[0];   // (16,1024,92) f32
    const float* ppts   = (const float*)d_in[1];   // (16,1024,2)  f32
    const int*   labels = (const int*)d_in[2];     // (16,128)     int
    const float* tpts   = (const float*)d_in[3];   // (16,128,2)   f32
    float* out = (float*)d_out;

    const size_t negBytes = (size_t)BS * Q * NCP * sizeof(float);   // ~6 MB
    const size_t cbtBytes = (size_t)BS * T * Q * sizeof(float);     // 8 MB
    float* negProb = (float*)d_ws;
    float* CbT = (ws_size >= negBytes + cbtBytes)
                     ? (float*)((char*)d_ws + negBytes) : nullptr;

    softmax_neg_kernel<<<(BS * Q) / 4, 128, 0, stream>>>(logits, negProb);
    cost_wmma_kernel<<<BS * (Q / 64), 128, 0, stream>>>(negProb, ppts, labels,
                                                        tpts, out, CbT);
    jv_kernel<<<BS, 256, 0, stream>>>(out, CbT, out);
}